// Image_encoder_61950608278169
// MI455X (gfx1250) — compile-verified
//
#include <hip/hip_runtime.h>
#include <math.h>

// N=512 images, D=768 d_mlp, C=2 classes
#define N_IMG 512
#define DMLP  768
#define NCLS  2
#define LROWP (DMLP + 4)   // LDS row pad: bank = (4m + d) % 64 -> conflict-free b64 reads

typedef __attribute__((ext_vector_type(2))) float v2f;
typedef __attribute__((ext_vector_type(4))) float v4f;
typedef __attribute__((ext_vector_type(8))) float v8f;

__device__ __forceinline__ float fast_tanh(float x) {
#if __has_builtin(__builtin_amdgcn_tanhf)
  return __builtin_amdgcn_tanhf(x);            // v_tanh_f32 (gfx1250 TRANS op)
#elif __has_builtin(__builtin_amdgcn_tanh_f32)
  return __builtin_amdgcn_tanh_f32(x);
#else
  return tanhf(x);
#endif
}

__device__ __forceinline__ v8f wmma4_f32(v2f a, v2f b, v8f c) {
  // (neg_a, A, neg_b, B, c_mod, C, reuse_a, reuse_b) -> v_wmma_f32_16x16x4_f32
  return __builtin_amdgcn_wmma_f32_16x16x4_f32(false, a, false, b, (short)0, c,
                                               false, false);
}

// ---- Kernel 0: zero-padded W2 (row 2 = zeros) so B-tile columns n>=2 read 0.
__global__ void build_w2z(const float* __restrict__ W2, float* __restrict__ W2Z) {
  int t = blockIdx.x * blockDim.x + threadIdx.x;
  if (t < 3 * DMLP) W2Z[t] = (t < NCLS * DMLP) ? W2[t] : 0.0f;
}

// ---- Kernel 1: LEFT = P @ W1[:, :D].T + b1 ; RIGHT = P @ W1[:, D:].T
// One wave per 16x16 output tile, K=768 in 192 WMMA steps of K=4.
__global__ __launch_bounds__(32) void gemm_lr(
    const float* __restrict__ P, const float* __restrict__ W1,
    const float* __restrict__ b1, float* __restrict__ LEFT,
    float* __restrict__ RIGHT) {
  const int lane = threadIdx.x & 31;
  const int m    = lane & 15;        // A: M row / B: N col (both = lane%16)
  const int half = lane >> 4;        // K sub-block select
  const int koff = half * 2;
  const int i0   = blockIdx.x * 16;
  const int o0   = blockIdx.y * 16;
  const int side = blockIdx.z;       // 0 = left (cols 0..D-1), 1 = right (cols D..2D-1)

  const float* arow = P  + (size_t)(i0 + m) * DMLP + koff;               // P[i][k]
  const float* brow = W1 + (size_t)(o0 + m) * (2 * DMLP) + side * DMLP + koff; // W1[o][k]

  v8f acc = {};
#pragma unroll 4
  for (int k0 = 0; k0 < DMLP; k0 += 4) {
    v2f a = *(const v2f*)(arow + k0);
    v2f b = *(const v2f*)(brow + k0);
    acc = wmma4_f32(a, b, acc);
  }
  // C/D layout: lanes 0-15: (M=r, N=lane); lanes 16-31: (M=8+r, N=lane-16)
  float bias = (side == 0) ? b1[o0 + m] : 0.0f;
  float* dst = (side == 0 ? LEFT : RIGHT) + (size_t)(i0 + half * 8) * DMLP + (o0 + m);
#pragma unroll
  for (int r = 0; r < 8; ++r) dst[(size_t)r * DMLP] = acc[r] + bias;
}

// ---- Kernel 2: logits[i,j,c] = sum_d tanh(LEFT[i,d] + RIGHT[j,d]) * W2[c,d] + b2[c]
// Block: 256 threads = 8 waves; owns one 16-row i-tile (LDS) x 32 j's.
// Each wave: 4 j's as 2 independent accumulator pairs (ILP, shared L/B loads).
// VALU does add+tanh only; d-contraction rides the matrix pipe (WMMA f32 16x16x4).
__global__ __launch_bounds__(256) void pair_kernel(
    const float* __restrict__ LEFT, const float* __restrict__ RIGHT,
    const float* __restrict__ W2Z, const float* __restrict__ b2,
    float* __restrict__ out) {
  __shared__ float ldsL[16 * LROWP];   // 49,408 B

  const int i0    = blockIdx.x * 16;
  const int jbase = blockIdx.y * 32;

  // Cooperative stage of the LEFT i-tile (16 x 768 f32) via b128 ds stores.
  for (int idx = threadIdx.x; idx < 16 * (DMLP / 4); idx += 256) {
    int mrow = idx / (DMLP / 4);
    int dv   = (idx % (DMLP / 4)) * 4;
    *(v4f*)&ldsL[mrow * LROWP + dv] =
        *(const v4f*)&LEFT[(size_t)(i0 + mrow) * DMLP + dv];
  }
  __syncthreads();

  const int lane = threadIdx.x & 31;
  const int wave = threadIdx.x >> 5;
  const int m    = lane & 15;
  const int half = lane >> 4;
  const int koff = half * 2;
  const int rn   = (m < NCLS) ? m : 2;             // padded-W2 row (row 2 = zeros)

  const float* Lrow = &ldsL[m * LROWP + koff];     // LDS, conflict-free
  const float* Brow = W2Z + rn * DMLP + koff;      // global, L0-resident (9 KB)

  for (int p = 0; p < 2; ++p) {
    const int j0 = jbase + wave * 4 + p * 2;
    const float* R0 = RIGHT + (size_t)j0 * DMLP + koff;
    const float* R1 = RIGHT + (size_t)(j0 + 1) * DMLP + koff;
    v8f acc0 = {};
    v8f acc1 = {};
#pragma unroll 4
    for (int d0 = 0; d0 < DMLP; d0 += 4) {
      v2f l  = *(const v2f*)(Lrow + d0);
      v2f b  = *(const v2f*)(Brow + d0);
      v2f r0 = *(const v2f*)(R0 + d0);
      v2f r1 = *(const v2f*)(R1 + d0);
      v2f a0, a1;
      a0.x = fast_tanh(l.x + r0.x);
      a0.y = fast_tanh(l.y + r0.y);
      a1.x = fast_tanh(l.x + r1.x);
      a1.y = fast_tanh(l.y + r1.y);
      acc0 = wmma4_f32(a0, b, acc0);
      acc1 = wmma4_f32(a1, b, acc1);
    }
    // Only D-matrix columns 0,1 are real classes: lanes 0,1 (M=0..7) and 16,17 (M=8..15).
    if (m < NCLS) {
      const float b2v  = b2[m];
      const int   mrow = i0 + half * 8;
#pragma unroll
      for (int r = 0; r < 8; ++r) {
        out[((size_t)(mrow + r) * N_IMG + j0) * NCLS + m]     = acc0[r] + b2v;
        out[((size_t)(mrow + r) * N_IMG + j0 + 1) * NCLS + m] = acc1[r] + b2v;
      }
    }
  }
}

extern "C" void kernel_launch(void* const* d_in, const int* in_sizes, int n_in,
                              void* d_out, int out_size, void* d_ws, size_t ws_size,
                              hipStream_t stream) {
  const float* P  = (const float*)d_in[0];  // [512, 768]
  const float* W1 = (const float*)d_in[1];  // [768, 1536]
  const float* b1 = (const float*)d_in[2];  // [768]
  const float* W2 = (const float*)d_in[3];  // [2, 768]
  const float* b2 = (const float*)d_in[4];  // [2]
  float* out = (float*)d_out;               // [512, 512, 2]

  float* ws    = (float*)d_ws;
  float* LEFT  = ws;                                  // 512*768 f32 (b1 folded in)
  float* RIGHT = ws + (size_t)N_IMG * DMLP;           // 512*768 f32
  float* W2Z   = ws + 2 * (size_t)N_IMG * DMLP;       // 3*768 f32 (row 2 zeros)

  build_w2z<<<dim3((3 * DMLP + 255) / 256), dim3(256), 0, stream>>>(W2, W2Z);
  gemm_lr<<<dim3(N_IMG / 16, DMLP / 16, 2), dim3(32), 0, stream>>>(P, W1, b1, LEFT, RIGHT);
  pair_kernel<<<dim3(N_IMG / 16, N_IMG / 32), dim3(256), 0, stream>>>(LEFT, RIGHT, W2Z, b2, out);
}